// CrossAttentionPoseRegression_26190710571147
// MI455X (gfx1250) — compile-verified
//
#include <hip/hip_runtime.h>
#include <math.h>

#define B_SZ   64
#define N_PTS  8192
#define FDIM   32
#define K_TOP  128

typedef __attribute__((ext_vector_type(2))) float v2f;
typedef __attribute__((ext_vector_type(8))) float v8f;

// workspace layout (float offsets)
#define OFF_SIM   0                         // B*N            = 524288
#define OFF_FEATS (OFF_SIM + B_SZ*N_PTS)    // B*768          = 49152
#define OFF_R     (OFF_FEATS + B_SZ*768)    // B*9
#define OFF_T     (OFF_R + B_SZ*9)          // B*3
#define OFF_A1    (OFF_T + B_SZ*3)          // B*256
#define OFF_A2    (OFF_A1 + B_SZ*256)       // B*128
#define OFF_A3    (OFF_A2 + B_SZ*128)       // B*64
#define OFF_D     (OFF_A3 + B_SZ*64)        // B*7

// ---------------------------------------------------------------- sim kernel
__global__ void sim_kernel(const float* __restrict__ hs, const float* __restrict__ ht,
                           float* __restrict__ sim) {
    int id = blockIdx.x * blockDim.x + threadIdx.x;
    if (id >= B_SZ * N_PTS) return;
    const float4* a = (const float4*)(hs + (size_t)id * FDIM);
    const float4* b = (const float4*)(ht + (size_t)id * FDIM);
    float acc = 0.f;
#pragma unroll
    for (int i = 0; i < FDIM / 4; ++i) {
        float4 x = a[i], y = b[i];
        acc += x.x * y.x + x.y * y.y + x.z * y.z + x.w * y.w;
    }
    sim[id] = acc;
}

// ------------------------------------------------------- top-k gather kernel
// Exact top-k via full bitonic sort of packed (descending-value, ascending-idx) keys.
__global__ void topk_kernel(const float* __restrict__ sim,
                            const float* __restrict__ xs, const float* __restrict__ xt,
                            float* __restrict__ feats) {
    __shared__ unsigned long long keys[N_PTS];  // 64 KB
    const int b = blockIdx.x, tid = threadIdx.x, nt = blockDim.x;
    for (int i = tid; i < N_PTS; i += nt) {
        unsigned u = __float_as_uint(sim[b * N_PTS + i]);
        u = (u & 0x80000000u) ? ~u : (u | 0x80000000u);    // monotonic ascending map
        keys[i] = ((unsigned long long)(~u) << 32) | (unsigned)i;
    }
    __syncthreads();
    for (int k = 2; k <= N_PTS; k <<= 1) {
        for (int j = k >> 1; j > 0; j >>= 1) {
            for (int i = tid; i < N_PTS; i += nt) {
                int ixj = i ^ j;
                if (ixj > i) {
                    unsigned long long a = keys[i], c = keys[ixj];
                    bool up = ((i & k) == 0);
                    if ((a > c) == up) { keys[i] = c; keys[ixj] = a; }
                }
            }
            __syncthreads();
        }
    }
    for (int kk = tid; kk < K_TOP; kk += nt) {
        int idx = (int)(keys[kk] & 0xFFFFFFFFull);
        const float* ps = xs + ((size_t)b * N_PTS + idx) * 3;
        const float* pt = xt + ((size_t)b * N_PTS + idx) * 3;
        float* f = feats + (size_t)b * (K_TOP * 6) + kk * 6;
        f[0] = ps[0]; f[1] = ps[1]; f[2] = ps[2];
        f[3] = pt[0]; f[4] = pt[1]; f[5] = pt[2];
    }
}

// ----------------------------------------------------------- Kabsch helpers
__device__ inline void jacobi3(float A[9], float V[9]) {
    for (int i = 0; i < 9; ++i) V[i] = 0.f;
    V[0] = V[4] = V[8] = 1.f;
    const int P[3] = {0, 0, 1}, Q[3] = {1, 2, 2};
    for (int sweep = 0; sweep < 24; ++sweep) {
        for (int pi = 0; pi < 3; ++pi) {
            int p = P[pi], q = Q[pi];
            float apq = A[p * 3 + q];
            if (fabsf(apq) < 1e-30f) continue;
            float app = A[p * 3 + p], aqq = A[q * 3 + q];
            float theta = 0.5f * (aqq - app) / apq;
            float tt = copysignf(1.f, theta) / (fabsf(theta) + sqrtf(theta * theta + 1.f));
            float c = 1.f / sqrtf(tt * tt + 1.f);
            float s = tt * c;
            for (int k = 0; k < 3; ++k) {
                float akp = A[k * 3 + p], akq = A[k * 3 + q];
                A[k * 3 + p] = c * akp - s * akq;
                A[k * 3 + q] = s * akp + c * akq;
            }
            for (int k = 0; k < 3; ++k) {
                float apk = A[p * 3 + k], aqk = A[q * 3 + k];
                A[p * 3 + k] = c * apk - s * aqk;
                A[q * 3 + k] = s * apk + c * aqk;
            }
            for (int k = 0; k < 3; ++k) {
                float vkp = V[k * 3 + p], vkq = V[k * 3 + q];
                V[k * 3 + p] = c * vkp - s * vkq;
                V[k * 3 + q] = s * vkp + c * vkq;
            }
        }
    }
}

__device__ inline float det3(const float R[9]) {
    return R[0] * (R[4] * R[8] - R[5] * R[7])
         - R[1] * (R[3] * R[8] - R[5] * R[6])
         + R[2] * (R[3] * R[7] - R[4] * R[6]);
}

// ------------------------------------------- masked-softmax moments + Kabsch
__global__ void kabsch_kernel(const float* __restrict__ sim,
                              const float* __restrict__ xs, const float* __restrict__ xt,
                              const int* __restrict__ labels,
                              float* __restrict__ Rout, float* __restrict__ tout) {
    __shared__ float red[256];
    __shared__ float out[17];
    const int b = blockIdx.x, tid = threadIdx.x, nt = 256;

    // masked max
    float m = -3.4e38f;
    for (int i = tid; i < N_PTS; i += nt) {
        float s = (labels[b * N_PTS + i] != 0) ? sim[b * N_PTS + i] : -1e9f;
        m = fmaxf(m, s);
    }
    red[tid] = m; __syncthreads();
    for (int s = nt / 2; s > 0; s >>= 1) {
        if (tid < s) red[tid] = fmaxf(red[tid], red[tid + s]);
        __syncthreads();
    }
    m = red[0]; __syncthreads();

    // moments: [0]=Se  [1..3]=Se*s  [4..6]=Se*t  [7..15]=Se*s_i*t_j  [16]=count
    float acc[17];
    for (int a = 0; a < 17; ++a) acc[a] = 0.f;
    for (int i = tid; i < N_PTS; i += nt) {
        if (labels[b * N_PTS + i] != 0) {
            float e = expf(sim[b * N_PTS + i] - m);
            const float* ps = xs + ((size_t)b * N_PTS + i) * 3;
            const float* pt = xt + ((size_t)b * N_PTS + i) * 3;
            acc[0] += e;
            acc[1] += e * ps[0]; acc[2] += e * ps[1]; acc[3] += e * ps[2];
            acc[4] += e * pt[0]; acc[5] += e * pt[1]; acc[6] += e * pt[2];
            for (int r = 0; r < 3; ++r)
                for (int c = 0; c < 3; ++c)
                    acc[7 + r * 3 + c] += e * ps[r] * pt[c];
            acc[16] += 1.f;
        }
    }
    for (int a = 0; a < 17; ++a) {
        red[tid] = acc[a]; __syncthreads();
        for (int s = nt / 2; s > 0; s >>= 1) {
            if (tid < s) red[tid] += red[tid + s];
            __syncthreads();
        }
        if (tid == 0) out[a] = red[0];
        __syncthreads();
    }

    if (tid == 0) {
        float R[9] = {1, 0, 0, 0, 1, 0, 0, 0, 1};
        float t[3] = {0, 0, 0};
        if (out[16] > 0.f) {
            float invW = 1.f / fmaxf(out[0], 1e-30f);
            float sc[3] = {out[1] * invW, out[2] * invW, out[3] * invW};
            float tc[3] = {out[4] * invW, out[5] * invW, out[6] * invW};
            float Hm[9];
            for (int r = 0; r < 3; ++r)
                for (int c = 0; c < 3; ++c)
                    Hm[r * 3 + c] = out[7 + r * 3 + c] * invW - sc[r] * tc[c];
            // A = H^T H ; eigen -> V (right singular vectors)
            float A[9];
            for (int i = 0; i < 3; ++i)
                for (int j = 0; j < 3; ++j) {
                    float su = 0.f;
                    for (int k = 0; k < 3; ++k) su += Hm[k * 3 + i] * Hm[k * 3 + j];
                    A[i * 3 + j] = su;
                }
            float V[9];
            jacobi3(A, V);
            float ev[3] = {A[0], A[4], A[8]};
            // sort eigenpairs descending
            for (int a = 0; a < 2; ++a)
                for (int bb = 0; bb < 2 - a; ++bb)
                    if (ev[bb] < ev[bb + 1]) {
                        float tmp = ev[bb]; ev[bb] = ev[bb + 1]; ev[bb + 1] = tmp;
                        for (int r = 0; r < 3; ++r) {
                            float tv2 = V[r * 3 + bb]; V[r * 3 + bb] = V[r * 3 + bb + 1]; V[r * 3 + bb + 1] = tv2;
                        }
                    }
            float sv[3] = {sqrtf(fmaxf(ev[0], 0.f)), sqrtf(fmaxf(ev[1], 0.f)), sqrtf(fmaxf(ev[2], 0.f))};
            // U = H V / sigma, with degeneracy fallbacks
            float HV[9];
            for (int i = 0; i < 3; ++i)
                for (int j = 0; j < 3; ++j) {
                    float su = 0.f;
                    for (int k = 0; k < 3; ++k) su += Hm[i * 3 + k] * V[k * 3 + j];
                    HV[i * 3 + j] = su;
                }
            float U[9] = {1, 0, 0, 0, 1, 0, 0, 0, 1};
            if (sv[0] > 1e-12f) {
                for (int j = 0; j < 3; ++j) {
                    if (sv[j] > 1e-12f) {
                        float n = 0.f;
                        for (int i = 0; i < 3; ++i) { U[i * 3 + j] = HV[i * 3 + j] / sv[j]; n += U[i * 3 + j] * U[i * 3 + j]; }
                        n = 1.f / fmaxf(sqrtf(n), 1e-20f);
                        for (int i = 0; i < 3; ++i) U[i * 3 + j] *= n;
                    } else if (j == 1) {
                        // pick vector perpendicular to column 0
                        float ax = fabsf(U[0]), ay = fabsf(U[3]), az = fabsf(U[6]);
                        float e0[3] = {0, 0, 0};
                        if (ax <= ay && ax <= az) e0[0] = 1.f;
                        else if (ay <= az) e0[1] = 1.f;
                        else e0[2] = 1.f;
                        float u0[3] = {U[0], U[3], U[6]};
                        float c[3] = {u0[1] * e0[2] - u0[2] * e0[1],
                                      u0[2] * e0[0] - u0[0] * e0[2],
                                      u0[0] * e0[1] - u0[1] * e0[0]};
                        float n = 1.f / fmaxf(sqrtf(c[0] * c[0] + c[1] * c[1] + c[2] * c[2]), 1e-20f);
                        U[1] = c[0] * n; U[4] = c[1] * n; U[7] = c[2] * n;
                    } else { // j == 2
                        U[2] = U[3] * U[7] - U[6] * U[4];
                        U[5] = U[6] * U[1] - U[0] * U[7];
                        U[8] = U[0] * U[4] - U[3] * U[1];
                    }
                }
            }
            // R = V U^T ; det sign fix on last column of V
            float R0[9];
            for (int i = 0; i < 3; ++i)
                for (int j = 0; j < 3; ++j) {
                    float su = 0.f;
                    for (int k = 0; k < 3; ++k) su += V[i * 3 + k] * U[j * 3 + k];
                    R0[i * 3 + j] = su;
                }
            if (det3(R0) < 0.f) {
                V[2] = -V[2]; V[5] = -V[5]; V[8] = -V[8];
                for (int i = 0; i < 3; ++i)
                    for (int j = 0; j < 3; ++j) {
                        float su = 0.f;
                        for (int k = 0; k < 3; ++k) su += V[i * 3 + k] * U[j * 3 + k];
                        R0[i * 3 + j] = su;
                    }
            }
            for (int i = 0; i < 9; ++i) R[i] = R0[i];
            for (int i = 0; i < 3; ++i)
                t[i] = tc[i] - (R[i * 3 + 0] * sc[0] + R[i * 3 + 1] * sc[1] + R[i * 3 + 2] * sc[2]);
        }
        for (int i = 0; i < 9; ++i) Rout[b * 9 + i] = R[i];
        for (int i = 0; i < 3; ++i) tout[b * 3 + i] = t[i];
    }
}

// ------------------------------------------------ WMMA f32 16x16x4 MLP layer
// A: M(=64) x K row-major ; Bm: K x Nvalid row-major ; C: 64 x Nvalid
__device__ inline void wmma_gemm(const float* __restrict__ A, const float* __restrict__ Bm,
                                 const float* __restrict__ bias, float* __restrict__ C,
                                 int K, int Nvalid, int tilesN, bool relu,
                                 int wave, int nWaves, int lane) {
    const int tilesM = 4;  // M = 64
    for (int t = wave; t < tilesM * tilesN; t += nWaves) {
        int tm = t / tilesN, tn = t % tilesN;
        v8f acc = {};
        int rowA = tm * 16 + (lane & 15);
        int colB = tn * 16 + (lane & 15);
        int kh = (lane >> 4) << 1;  // lanes 16-31 hold K+2/K+3
        bool bok = colB < Nvalid;
        for (int k0 = 0; k0 < K; k0 += 4) {
            int ka = k0 + kh;
            v2f av, bv;
            av.x = A[rowA * K + ka];
            av.y = A[rowA * K + ka + 1];
            bv.x = bok ? Bm[ka * Nvalid + colB] : 0.f;
            bv.y = bok ? Bm[(ka + 1) * Nvalid + colB] : 0.f;
            acc = __builtin_amdgcn_wmma_f32_16x16x4_f32(
                false, av, false, bv, (short)0, acc, false, false);
        }
        if (bok) {
            float bb = bias[colB];
            int rbase = tm * 16 + ((lane >> 4) << 3);  // D: VGPR v -> M=v (lanes<16) / v+8
#pragma unroll
            for (int v = 0; v < 8; ++v) {
                float val = acc[v] + bb;
                if (relu) val = fmaxf(val, 0.f);
                C[(rbase + v) * Nvalid + colB] = val;
            }
        }
    }
}

__global__ void __launch_bounds__(256) mlp_kernel(
    const float* __restrict__ feats,
    const float* __restrict__ W1, const float* __restrict__ b1,
    const float* __restrict__ W2, const float* __restrict__ b2,
    const float* __restrict__ W3, const float* __restrict__ b3,
    const float* __restrict__ W4, const float* __restrict__ b4,
    const float* __restrict__ Rm, const float* __restrict__ tv,
    float* __restrict__ act1, float* __restrict__ act2, float* __restrict__ act3,
    float* __restrict__ delta, float* __restrict__ outp) {
    const int tid = threadIdx.x;
    const int lane = tid & 31, wave = tid >> 5, nW = 8;

    wmma_gemm(feats, W1, b1, act1, 768, 256, 16, true, wave, nW, lane);
    __threadfence(); __syncthreads();
    wmma_gemm(act1, W2, b2, act2, 256, 128, 8, true, wave, nW, lane);
    __threadfence(); __syncthreads();
    wmma_gemm(act2, W3, b3, act3, 128, 64, 4, true, wave, nW, lane);
    __threadfence(); __syncthreads();
    wmma_gemm(act3, W4, b4, delta, 64, 7, 1, false, wave, nW, lane);
    __threadfence(); __syncthreads();

    if (tid < B_SZ) {
        const int b = tid;
        const float* d = delta + b * 7;
        float qn = sqrtf(d[0] * d[0] + d[1] * d[1] + d[2] * d[2] + d[3] * d[3]);
        float inv = 1.f / fmaxf(qn, 1e-12f);
        float qx = d[0] * inv, qy = d[1] * inv, qz = d[2] * inv, qw = d[3] * inv;
        float dR[9] = {
            1.f - 2.f * (qy * qy + qz * qz), 2.f * (qx * qy - qz * qw), 2.f * (qx * qz + qy * qw),
            2.f * (qx * qy + qz * qw), 1.f - 2.f * (qx * qx + qz * qz), 2.f * (qy * qz - qx * qw),
            2.f * (qx * qz - qy * qw), 2.f * (qy * qz + qx * qw), 1.f - 2.f * (qx * qx + qy * qy)};
        float R[9];
        for (int i = 0; i < 3; ++i)
            for (int j = 0; j < 3; ++j) {
                float su = 0.f;
                for (int k = 0; k < 3; ++k) su += dR[i * 3 + k] * Rm[b * 9 + k * 3 + j];
                R[i * 3 + j] = su;
            }
        float rt[3] = {tv[b * 3 + 0] + d[4], tv[b * 3 + 1] + d[5], tv[b * 3 + 2] + d[6]};
        float R00 = R[0], R01 = R[1], R02 = R[2];
        float R10 = R[3], R11 = R[4], R12 = R[5];
        float R20 = R[6], R21 = R[7], R22 = R[8];
        float tr4 = R00 + R11 + R22 + 1.f;
        float q0, q1, q2, q3;
        if (tr4 > 0.f) {
            float S = sqrtf(fmaxf(tr4 + 1.f, 1e-9f)) * 2.f;
            q0 = 0.25f * S; q1 = (R21 - R12) / S; q2 = (R02 - R20) / S; q3 = (R10 - R01) / S;
        } else if (R00 > R11 && R00 > R22) {
            float S = sqrtf(fmaxf(1.f + R00 - R11 - R22, 1e-9f)) * 2.f;
            q0 = (R21 - R12) / S; q1 = 0.25f * S; q2 = (R01 + R10) / S; q3 = (R02 + R20) / S;
        } else if (R11 > R22) {
            float S = sqrtf(fmaxf(1.f + R11 - R00 - R22, 1e-9f)) * 2.f;
            q0 = (R02 - R20) / S; q1 = (R01 + R10) / S; q2 = 0.25f * S; q3 = (R12 + R21) / S;
        } else {
            float S = sqrtf(fmaxf(1.f + R22 - R00 - R11, 1e-9f)) * 2.f;
            q0 = (R10 - R01) / S; q1 = (R02 + R20) / S; q2 = (R12 + R21) / S; q3 = 0.25f * S;
        }
        float n2 = 1.f / fmaxf(sqrtf(q0 * q0 + q1 * q1 + q2 * q2 + q3 * q3), 1e-12f);
        outp[b * 4 + 0] = q0 * n2;
        outp[b * 4 + 1] = q1 * n2;
        outp[b * 4 + 2] = q2 * n2;
        outp[b * 4 + 3] = q3 * n2;
        outp[B_SZ * 4 + b * 3 + 0] = rt[0];
        outp[B_SZ * 4 + b * 3 + 1] = rt[1];
        outp[B_SZ * 4 + b * 3 + 2] = rt[2];
    }
}

// ----------------------------------------------------------------- launcher
extern "C" void kernel_launch(void* const* d_in, const int* in_sizes, int n_in,
                              void* d_out, int out_size, void* d_ws, size_t ws_size,
                              hipStream_t stream) {
    (void)in_sizes; (void)n_in; (void)out_size; (void)ws_size;
    const float* h_src = (const float*)d_in[0];
    const float* x_src = (const float*)d_in[1];
    const float* h_tgt = (const float*)d_in[4];
    const float* x_tgt = (const float*)d_in[5];
    const int*   labels = (const int*)d_in[9];
    const float* W1 = (const float*)d_in[10];
    const float* b1 = (const float*)d_in[11];
    const float* W2 = (const float*)d_in[12];
    const float* b2 = (const float*)d_in[13];
    const float* W3 = (const float*)d_in[14];
    const float* b3 = (const float*)d_in[15];
    const float* W4 = (const float*)d_in[16];
    const float* b4 = (const float*)d_in[17];

    float* ws    = (float*)d_ws;
    float* sim   = ws + OFF_SIM;
    float* feats = ws + OFF_FEATS;
    float* Rm    = ws + OFF_R;
    float* tv    = ws + OFF_T;
    float* act1  = ws + OFF_A1;
    float* act2  = ws + OFF_A2;
    float* act3  = ws + OFF_A3;
    float* delta = ws + OFF_D;

    sim_kernel<<<(B_SZ * N_PTS) / 256, 256, 0, stream>>>(h_src, h_tgt, sim);
    topk_kernel<<<B_SZ, 512, 0, stream>>>(sim, x_src, x_tgt, feats);
    kabsch_kernel<<<B_SZ, 256, 0, stream>>>(sim, x_src, x_tgt, labels, Rm, tv);
    mlp_kernel<<<1, 256, 0, stream>>>(feats, W1, b1, W2, b2, W3, b3, W4, b4,
                                      Rm, tv, act1, act2, act3, delta, (float*)d_out);
}